// MultiLoss_KLD_6579889897515
// MI455X (gfx1250) — compile-verified
//
#include <hip/hip_runtime.h>
#include <math.h>

#define NUM_ENC 10
#define NCOLS   50
#define EPSF    1e-10f
#define TPB     128
#define WPB     4     // waves per block (wave32)

// ---- workspace layout (float indices) ----
#define WS_HIST   0     // 200 floats: [group(2)][dim(10)][bin(10)]
#define WS_CNT_S  200
#define WS_CNT_M  201
#define WS_MSE    202
#define WS_CE     203
#define WS_MINU   224   // 10 uints (monotone-encoded float min)
#define WS_MAXU   234   // 10 uints (monotone-encoded float max)

typedef __attribute__((ext_vector_type(2))) float v2f;
typedef __attribute__((ext_vector_type(8))) float v8f;

// order-preserving float<->uint map (exact min/max via integer atomics)
__device__ __forceinline__ unsigned f2mono(float f) {
  unsigned b = __float_as_uint(f);
  return (b & 0x80000000u) ? ~b : (b | 0x80000000u);
}
__device__ __forceinline__ float mono2f(unsigned u) {
  unsigned b = (u & 0x80000000u) ? (u & 0x7fffffffu) : ~u;
  return __uint_as_float(b);
}

// Full-precision wave32 sum using V_WMMA_F32_16X16X4_F32.
// A[m][0]=v(lane m<16), A[m][2]=v(lane m+16), other K slots 0; B = all ones
// (layout-independent).  D[m][n] = v[m] + v[m+16]; summing the 8 D VGPRs
// in-lane gives half-sums in lane halves; one xor-16 shuffle finishes.
__device__ __forceinline__ float wave_sum_wmma(float v) {
  v2f a; a[0] = v;    a[1] = 0.0f;
  v2f b; b[0] = 1.0f; b[1] = 1.0f;
  v8f c = {};
  c = __builtin_amdgcn_wmma_f32_16x16x4_f32(false, a, false, b, (short)0, c,
                                            false, false);
  float s = ((c[0] + c[1]) + (c[2] + c[3])) + ((c[4] + c[5]) + (c[6] + c[7]));
  s += __shfl_xor(s, 16, 32);
  return s;
}

// ---------------- kernel 0: init workspace ----------------
__global__ void k_init_ws(float* __restrict__ W) {
  int t = threadIdx.x;
  if (t < 204) W[t] = 0.0f;
  unsigned* WU = (unsigned*)W;
  if (t >= WS_MINU && t < WS_MINU + 10) WU[t] = 0xFFFFFFFFu;
  if (t >= WS_MAXU && t < WS_MAXU + 10) WU[t] = 0u;
}

// ---------------- kernel 1: per-column min/max over enc ----------------
__global__ __launch_bounds__(256) void k_minmax(const float* __restrict__ enc,
                                                unsigned* __restrict__ WU,
                                                int B) {
  float mn[NUM_ENC], mx[NUM_ENC];
#pragma unroll
  for (int d = 0; d < NUM_ENC; ++d) { mn[d] = INFINITY; mx[d] = -INFINITY; }
  const float2* e2 = (const float2*)enc;
  int stride = gridDim.x * blockDim.x;
  for (int r = blockIdx.x * blockDim.x + threadIdx.x; r < B; r += stride) {
#pragma unroll
    for (int jv = 0; jv < 5; ++jv) {
      float2 v = e2[r * 5 + jv];
      int d = jv * 2;
      mn[d]     = fminf(mn[d], v.x);     mx[d]     = fmaxf(mx[d], v.x);
      mn[d + 1] = fminf(mn[d + 1], v.y); mx[d + 1] = fmaxf(mx[d + 1], v.y);
    }
  }
  __shared__ unsigned smin[NUM_ENC], smax[NUM_ENC];
  if (threadIdx.x < NUM_ENC) { smin[threadIdx.x] = 0xFFFFFFFFu; smax[threadIdx.x] = 0u; }
  __syncthreads();
#pragma unroll
  for (int d = 0; d < NUM_ENC; ++d) {
    atomicMin(&smin[d], f2mono(mn[d]));
    atomicMax(&smax[d], f2mono(mx[d]));
  }
  __syncthreads();
  if (threadIdx.x < NUM_ENC) {
    atomicMin(&WU[WS_MINU + threadIdx.x], smin[threadIdx.x]);
    atomicMax(&WU[WS_MAXU + threadIdx.x], smax[threadIdx.x]);
  }
}

// ---------------- kernel 2: main streaming pass ----------------
__global__ __launch_bounds__(TPB) void k_main(const float* __restrict__ enc,
                                              const float* __restrict__ dec,
                                              const float* __restrict__ tru,
                                              const float* __restrict__ lab,
                                              float* __restrict__ W, int B) {
  __shared__ float s_dec[WPB][32 * NCOLS];   // 25.6 KB
  __shared__ float s_tru[WPB][32 * NCOLS];   // 25.6 KB
  __shared__ float s_hist[200];

  const int tid  = threadIdx.x;
  const int w    = tid >> 5;
  const int lane = tid & 31;

  for (int i = tid; i < 200; i += TPB) s_hist[i] = 0.0f;
  __syncthreads();

  // decode exact global min/max and bin widths
  const unsigned* WU = (const unsigned*)W;
  float mnv[NUM_ENC], wdt[NUM_ENC];
#pragma unroll
  for (int d = 0; d < NUM_ENC; ++d) {
    float mn  = mono2f(WU[WS_MINU + d]);
    float mxv = mono2f(WU[WS_MAXU + d]);
    mnv[d] = mn;
    wdt[d] = fmaxf(mxv - mn, EPSF);
  }

  const int bs[9] = {7, 19, 21, 25, 27, 29, 31, 34, 38};
  const int be[9] = {19, 21, 25, 27, 29, 31, 34, 38, 50};

  float acc_mse = 0.f, acc_ce = 0.f, acc_cs = 0.f, acc_cm = 0.f;

  const int NT = (B + 31) >> 5;                       // 32-row tiles
  const unsigned ldsD = (unsigned)(size_t)&s_dec[w][0];
  const unsigned ldsT = (unsigned)(size_t)&s_tru[w][0];
  const float2* e2 = (const float2*)enc;

  for (int tile = blockIdx.x * WPB + w; tile < NT; tile += gridDim.x * WPB) {
    // ---- async stage: 32 rows x 50 f32 (6400 B) of dec & true -> LDS ----
    // 12 x b128 (6144 B) + 1 x b64 tail (256 B); both 16B-aligned since
    // gbase = tile*6400 and hipMalloc bases are 256B-aligned.
    const unsigned gbase = (unsigned)tile * (32u * NCOLS * 4u);  // bytes
#pragma unroll
    for (int i = 0; i < 12; ++i) {
      unsigned loff = (unsigned)((i * 32 + lane) * 16);
      unsigned goff = gbase + loff;
      asm volatile("global_load_async_to_lds_b128 %0, %1, %2 offset:0"
                   :: "v"(ldsD + loff), "v"(goff), "s"(dec) : "memory");
      asm volatile("global_load_async_to_lds_b128 %0, %1, %2 offset:0"
                   :: "v"(ldsT + loff), "v"(goff), "s"(tru) : "memory");
    }
    {
      unsigned loff = 6144u + (unsigned)(lane * 8);
      unsigned goff = gbase + loff;
      asm volatile("global_load_async_to_lds_b64 %0, %1, %2 offset:0"
                   :: "v"(ldsD + loff), "v"(goff), "s"(dec) : "memory");
      asm volatile("global_load_async_to_lds_b64 %0, %1, %2 offset:0"
                   :: "v"(ldsT + loff), "v"(goff), "s"(tru) : "memory");
    }
    asm volatile("s_wait_asynccnt 0" ::: "memory");     // wave-private tile

    int r = tile * 32 + lane;
    if (r < B) {
      const float* drow = &s_dec[w][lane * NCOLS];
      const float* trow = &s_tru[w][lane * NCOLS];

      // MSE over cols 0..6
      float m7 = 0.f;
#pragma unroll
      for (int j = 0; j < 7; ++j) {
        float d = drow[j] - trow[j];
        m7 += d * d;
      }
      acc_mse += m7;

      // CE over 9 one-hot blocks:  lse(x) - <onehot, x>
      float ceb = 0.f;
#pragma unroll
      for (int b = 0; b < 9; ++b) {
        float mxv = -INFINITY, xl = 0.f;
        for (int j = bs[b]; j < be[b]; ++j) {
          float x = drow[j];
          mxv = fmaxf(mxv, x);
          xl += trow[j] * x;
        }
        float se = 0.f;
        for (int j = bs[b]; j < be[b]; ++j) se += expf(drow[j] - mxv);
        ceb += mxv + logf(se) - xl;
      }
      acc_ce += ceb;

      // masked histograms over enc
      float g  = lab[r * 3 + 1];
      bool isS = (g == 0.0f), isM = (g == 1.0f);
      acc_cs += isS ? 1.0f : 0.0f;
      acc_cm += isM ? 1.0f : 0.0f;
      if (isS || isM) {
        int goff2 = isM ? 100 : 0;
#pragma unroll
        for (int jv = 0; jv < 5; ++jv) {
          float2 v = e2[r * 5 + jv];
#pragma unroll
          for (int k = 0; k < 2; ++k) {
            int d   = jv * 2 + k;
            float x = (k == 0) ? v.x : v.y;
            float bf = floorf((x - mnv[d]) / wdt[d] * 10.0f);
            int bi = (int)fminf(fmaxf(bf, 0.0f), 9.0f);
            atomicAdd(&s_hist[goff2 + d * 10 + bi], 1.0f);   // ds_add_f32
          }
        }
      }
    }
  }

  // wave reductions via v_wmma_f32_16x16x4_f32 (EXEC all-ones here)
  float tm = wave_sum_wmma(acc_mse);
  float tc = wave_sum_wmma(acc_ce);
  float ts = wave_sum_wmma(acc_cs);
  float tr = wave_sum_wmma(acc_cm);
  if (lane == 0) {
    atomicAdd(&W[WS_MSE],   tm);
    atomicAdd(&W[WS_CE],    tc);
    atomicAdd(&W[WS_CNT_S], ts);
    atomicAdd(&W[WS_CNT_M], tr);
  }

  __syncthreads();
  for (int i = tid; i < 200; i += TPB)
    atomicAdd(&W[WS_HIST + i], s_hist[i]);
}

// ---------------- kernel 3: finalize (tiny) ----------------
__global__ void k_final(const float* __restrict__ W, float* __restrict__ out,
                        int B) {
  if (threadIdx.x == 0 && blockIdx.x == 0) {
    float invB = 1.0f / (float)B;
    float mse = W[WS_MSE] * invB;
    float ce  = W[WS_CE]  * invB;
    float dsn = fmaxf(W[WS_CNT_S], 1.0f);
    float dmn = fmaxf(W[WS_CNT_M], 1.0f);
    float kld = 0.0f;
    for (int i = 0; i < 100; ++i) {
      float p = W[WS_HIST + i] / dsn;
      float q = W[WS_HIST + 100 + i] / dmn;
      if (p > 0.0f) kld += p * logf(p / fmaxf(q, EPSF));
    }
    out[0] = 0.5f * (mse + ce) + 0.5f * kld;   // ALPHA = 0.5
    out[1] = mse;
    out[2] = ce;
    out[3] = 0.5f * kld;
  }
}

extern "C" void kernel_launch(void* const* d_in, const int* in_sizes, int n_in,
                              void* d_out, int out_size, void* d_ws,
                              size_t ws_size, hipStream_t stream) {
  const float* enc = (const float*)d_in[0];   // (B,10)
  const float* dec = (const float*)d_in[1];   // (B,50)
  const float* tru = (const float*)d_in[2];   // (B,50)
  const float* lab = (const float*)d_in[3];   // (B,3)
  const int B = in_sizes[0] / NUM_ENC;

  float* W   = (float*)d_ws;
  float* out = (float*)d_out;

  k_init_ws<<<1, 256, 0, stream>>>(W);
  k_minmax<<<2048, 256, 0, stream>>>(enc, (unsigned*)W, B);
  k_main<<<2048, TPB, 0, stream>>>(enc, dec, tru, lab, W, B);
  k_final<<<1, 32, 0, stream>>>(W, out, B);
}